// R4D_83073257439376
// MI455X (gfx1250) — compile-verified
//
#include <hip/hip_runtime.h>
#include <stdint.h>

typedef __attribute__((ext_vector_type(2))) float v2f;
typedef __attribute__((ext_vector_type(8))) float v8f;

#define HW_    196
#define THW    784                    /* t folded into n: 4*196 */
#define CTHW   802816                 /* 1024*784 */
#define K_CH   32
#define APITCH 136                    /* 128 + 8: K+2 plane lands 16 banks away */
#define BPITCH 120                    /* 112 + 8: K+2 plane lands 48 banks away */
#define A_SZ   (K_CH * APITCH)        /* 4352 floats */
#define B_SZ   (K_CH * BPITCH)        /* 3840 floats */
#define BUF_SZ (A_SZ + B_SZ)          /* 8192 floats; x2 buffers = 64KB LDS */

// ---- CDNA5 async global->LDS copies (ASYNCcnt) -----------------------------
__device__ __forceinline__ void async_ld_b128(uint32_t ldsoff, const void* base,
                                              uint32_t voff) {
  asm volatile("global_load_async_to_lds_b128 %0, %1, %2"
               :: "v"(ldsoff), "v"(voff), "s"(base) : "memory");
}
__device__ __forceinline__ void async_ld_b64(uint32_t ldsoff, const void* base,
                                             uint32_t voff) {
  asm volatile("global_load_async_to_lds_b64 %0, %1, %2"
               :: "v"(ldsoff), "v"(voff), "s"(base) : "memory");
}
__device__ __forceinline__ void wait_async0() {
  asm volatile("s_wait_asynccnt 0" ::: "memory");
}

// ---------------------------------------------------------------------------
// Repack W (co, ci, 3, 3) -> Wt[tap][ci][co] via LDS transpose (coalesced
// reads AND writes; 289 pitch -> conflict-free banks).
// ---------------------------------------------------------------------------
__global__ void __launch_bounds__(256) repack_w_kernel(
    const float* __restrict__ W, float* __restrict__ Wt) {
  __shared__ float tile[32 * 289];
  const int co0 = (blockIdx.x & 31) * 32;
  const int ci0 = (blockIdx.x >> 5) * 32;
  const int tid = threadIdx.x;
#pragma unroll
  for (int i = 0; i < 36; ++i) {
    int idx = i * 256 + tid;               // 0..9215
    int row = idx / 288;                   // co_local
    int col = idx - row * 288;             // ci_local*9 + tap
    tile[row * 289 + col] = W[(long)(co0 + row) * 9216 + ci0 * 9 + col];
  }
  __syncthreads();
#pragma unroll
  for (int i = 0; i < 36; ++i) {
    int u    = i * 256 + tid;              // tap*1024 + ci_l*32 + co_l
    int co_l = u & 31;
    int ci_l = (u >> 5) & 31;
    int tap  = u >> 10;
    Wt[(long)(tap * 1024 + ci0 + ci_l) * 1024 + co0 + co_l] =
        tile[co_l * 289 + ci_l * 9 + tap];
  }
}

// ---------------------------------------------------------------------------
// Implicit-GEMM conv, t folded into N (784 = 7*112, no padding waste).
// Grid: 8 (b2,s) x 8 M-tiles(128) x 7 N-tiles(112) = 448 WGs.
// Block: 8 waves; wave w = rows [m0+16w, +16), 7 accumulators of 16x16.
// Double-buffered async-to-LDS staging of A (W tile) and B (x tile, with
// kt-shift edge columns zero-filled at stage time).
// ---------------------------------------------------------------------------
__global__ void __launch_bounds__(256) conv_wmma_kernel(
    const float* __restrict__ X, const float* __restrict__ Wt,
    float* __restrict__ Y) {
  __shared__ float lds[2 * BUF_SZ];

  const int gid   = blockIdx.x;
  const int nt    = gid % 7;
  const int rest  = gid / 7;
  const int mt    = rest & 7;
  const int combo = rest >> 3;             // 0..7
  const int s  = combo & 3;
  const int b2 = combo >> 2;
  const int m_base = mt * 128;
  const int n_base = nt * 112;

  const int tid  = threadIdx.x;
  const int wv   = tid >> 5;
  const int lane = tid & 31;
  const int half = lane >> 4;              // K-pair select (A lanes 16-31: K=2,3)
  const int l15  = lane & 15;

  const int ks0   = (s == 0) ? 1 : 0;      // valid ks range (padding along s)
  const int n_ks  = (s == 0 || s == 3) ? 2 : 3;
  const int total = n_ks * 3 * (1024 / K_CH);   // passes * K-chunks

  const uint32_t lds_base = (uint32_t)(uintptr_t)(&lds[0]);
  const int srow = tid >> 3;               // staging: K row 0..31
  const int scs  = tid & 7;                // staging: column slot

  auto stage = [&](int u, int buf) {
    const int k0   = (u & 31) * K_CH;
    const int pass = u >> 5;
    const int ksi  = pass / 3;
    const int kt   = pass - ksi * 3;
    const int ks   = ks0 + ksi;
    const int sp   = s + ks - 1;
    const int tap  = ks * 3 + kt;
    const int shift = (kt - 1) * HW_;      // kt tap = column shift in folded n
    // A: Wt[tap][k0+srow][m_base .. +128): 16 floats/thread as 4 x b128
    uint32_t ag = (uint32_t)((tap << 20) + (k0 + srow) * 1024 + m_base + scs * 16) * 4u;
    uint32_t al = lds_base + (uint32_t)(buf * BUF_SZ + srow * APITCH + scs * 16) * 4u;
#pragma unroll
    for (int q = 0; q < 4; ++q)
      async_ld_b128(al + q * 16u, Wt, ag + q * 16u);
    // B: x[b2,sp][k0+srow][n_base+shift ..]: 14 floats/thread as 7 x b64,
    // zero-filling columns masked by the t-padding.
    const int xrow  = (b2 * 4 + sp) * CTHW + (k0 + srow) * THW;
    const int lbase = buf * BUF_SZ + A_SZ + srow * BPITCH;
#pragma unroll
    for (int q = 0; q < 7; ++q) {
      int c = scs * 14 + q * 2;
      int g = n_base + c + shift;          // even; b64 never straddles the edge
      if ((unsigned)g < (unsigned)THW) {
        async_ld_b64(lds_base + (uint32_t)(lbase + c) * 4u, X,
                     (uint32_t)(xrow + g) * 4u);
      } else {
        lds[lbase + c]     = 0.0f;
        lds[lbase + c + 1] = 0.0f;
      }
    }
  };

  v8f acc[7] = {};

  stage(0, 0);
  wait_async0();
  __syncthreads();

  int cur = 0;
  for (int u = 0; u < total; ++u) {
    if (u + 1 < total) stage(u + 1, cur ^ 1);       // overlap DMA with WMMA
    const float* Ab = &lds[cur * BUF_SZ];
    const float* Bb = &lds[cur * BUF_SZ + A_SZ];
#pragma unroll
    for (int kk = 0; kk < K_CH; kk += 4) {
      const int bk = kk + 2 * half;
      const float* ap = Ab + bk * APITCH + wv * 16 + l15;
      v2f a;
      a.x = ap[0];
      a.y = ap[APITCH];
      const float* bp = Bb + bk * BPITCH + l15;
#pragma unroll
      for (int j = 0; j < 7; ++j) {
        v2f b;
        b.x = bp[j * 16];
        b.y = bp[BPITCH + j * 16];
        acc[j] = __builtin_amdgcn_wmma_f32_16x16x4_f32(
            false, a, false, b, (short)0, acc[j], false, false);
      }
    }
    wait_async0();          // next chunk fully in LDS
    __syncthreads();        // all waves done reading/staging this round
    cur ^= 1;
  }

  // C/D layout: VGPR r -> row half*8 + r; lane[3:0] -> column. Fused residual.
  const int obase = (b2 * 4 + s) * CTHW;
  const int row0  = m_base + wv * 16 + half * 8;
#pragma unroll
  for (int j = 0; j < 7; ++j) {
    const int col = n_base + j * 16 + l15;
#pragma unroll
    for (int r = 0; r < 8; ++r) {
      int idx = obase + (row0 + r) * THW + col;
      Y[idx] = acc[j][r] + X[idx];
    }
  }
}

extern "C" void kernel_launch(void* const* d_in, const int* in_sizes, int n_in,
                              void* d_out, int out_size, void* d_ws, size_t ws_size,
                              hipStream_t stream) {
  const float* X = (const float*)d_in[0];   // (8,1024,4,14,14) fp32
  const float* W = (const float*)d_in[1];   // (1024,1024,3,3)  fp32
  float* Y  = (float*)d_out;
  float* Wt = (float*)d_ws;                 // needs 9*1024*1024*4 = 37,748,736 B

  (void)in_sizes; (void)n_in; (void)out_size; (void)ws_size;

  repack_w_kernel<<<32 * 32, 256, 0, stream>>>(W, Wt);
  conv_wmma_kernel<<<8 * 8 * 7, 256, 0, stream>>>(X, Wt, Y);
}